// VertexUpdate_69621419868748
// MI455X (gfx1250) — compile-verified
//
#include <hip/hip_runtime.h>

typedef __attribute__((ext_vector_type(2))) float v2f;
typedef __attribute__((ext_vector_type(8))) float v8f;

// ---------------- Stage 0: zero the aggregation scratch ----------------
__global__ void zero_agg_kernel(float* __restrict__ agg, int n) {
  int i = blockIdx.x * blockDim.x + threadIdx.x;
  if (i < n) agg[i] = 0.0f;
}

// ---------------- Stage 1: edge scatter (segment_sum) ------------------
// agg[dst][c] += edge_attr[e][c]  via f32 L2 atomics (accumulator is 6MB,
// fully L2-resident; the 256MB of streaming reads set the runtime floor).
__global__ void edge_scatter_kernel(const float* __restrict__ edge_attr,
                                    const int* __restrict__ edge_dst,
                                    float* __restrict__ agg, int nEdges) {
  int e = blockIdx.x * blockDim.x + threadIdx.x;
  if (e >= nEdges) return;
  int dst = edge_dst[e];
  float a0 = edge_attr[3 * e + 0];
  float a1 = edge_attr[3 * e + 1];
  float a2 = edge_attr[3 * e + 2];
  atomicAdd(&agg[3 * dst + 0], a0);
  atomicAdd(&agg[3 * dst + 1], a1);
  atomicAdd(&agg[3 * dst + 2], a2);
}

// ---------------- Stage 2: per-node MLP via fp32 WMMA -------------------
// One wave32 handles 16 nodes (N dim of the 16x16x4 tiles).
// Layout conventions (ISA 7.12.2, 32-bit):
//   A 16x4:  lane<16: (K=0,K=1) in (v0,v1); lane>=16: (K=2,K=3)
//   B 4x16:  same K striping across lane halves, N = lane&15
//   C/D:     d[v], lane l -> row M = v + 8*(l>>4), col N = l&15
__global__ __launch_bounds__(256) void mlp_wmma_kernel(
    const float* __restrict__ vertex, const float* __restrict__ agg,
    const float* __restrict__ W1, const float* __restrict__ b1,
    const float* __restrict__ W2, const float* __restrict__ b2,
    const float* __restrict__ W3, const float* __restrict__ b3,
    float* __restrict__ out, int nNodes) {
  __shared__ float h1s[8][64 * 16];  // layer-1 activations, zero-padded rows 50..63
  __shared__ float h2s[8][32 * 16];  // layer-2 activations, zero-padded rows 20..31

  const int lane = threadIdx.x & 31;
  const int wave = threadIdx.x >> 5;
  const int half = lane >> 4;   // which K pair this lane supplies
  const int col  = lane & 15;   // node slot / A row within tile
  const int gbase = (blockIdx.x * 8 + wave) * 16;

  int n = gbase + col;
  if (n >= nNodes) n = nNodes - 1;  // clamp for loads; store is guarded later

  // x = [v0, v1, agg0, agg1, agg2]
  const float x0 = vertex[2 * n + 0];
  const float x1 = vertex[2 * n + 1];
  const float x2 = agg[3 * n + 0];
  const float x3 = agg[3 * n + 1];
  const float x4 = agg[3 * n + 2];

  // Layer-1 B tiles built directly in registers (K padded 5 -> 8).
  v2f bt0, bt1;
  bt0.x = half ? x2 : x0;
  bt0.y = half ? x3 : x1;
  bt1.x = half ? 0.0f : x4;
  bt1.y = 0.0f;

  float* l1 = h1s[wave];

  // ---- Layer 1: 50x5, 4 M-tiles x 2 K-tiles ----
#pragma unroll
  for (int m = 0; m < 4; ++m) {
    v8f d = {};
#pragma unroll
    for (int k = 0; k < 2; ++k) {
      const int r  = 16 * m + col;
      const int kb = 4 * k + 2 * half;
      v2f a;
      a.x = (r < 50 && kb     < 5) ? W1[r * 5 + kb]     : 0.0f;
      a.y = (r < 50 && kb + 1 < 5) ? W1[r * 5 + kb + 1] : 0.0f;
      v2f b = (k == 0) ? bt0 : bt1;
      d = __builtin_amdgcn_wmma_f32_16x16x4_f32(false, a, false, b,
                                                (short)0, d, false, false);
    }
#pragma unroll
    for (int v = 0; v < 8; ++v) {
      const int feat = 16 * m + v + 8 * half;
      float val = d[v] + (feat < 50 ? b1[feat] : 0.0f);
      l1[feat * 16 + col] = val > 0.0f ? val : 0.0f;  // ReLU; pad rows -> 0
    }
  }
  __syncthreads();

  float* l2 = h2s[wave];

  // ---- Layer 2: 20x50, 2 M-tiles x 13 K-tiles (K padded 50 -> 52) ----
#pragma unroll
  for (int m = 0; m < 2; ++m) {
    v8f d = {};
#pragma unroll
    for (int k = 0; k < 13; ++k) {
      const int r  = 16 * m + col;
      const int kb = 4 * k + 2 * half;
      v2f a, b;
      a.x = (r < 20 && kb     < 50) ? W2[r * 50 + kb]     : 0.0f;
      a.y = (r < 20 && kb + 1 < 50) ? W2[r * 50 + kb + 1] : 0.0f;
      b.x = l1[(kb)     * 16 + col];  // rows 50..63 are zero-padded
      b.y = l1[(kb + 1) * 16 + col];
      d = __builtin_amdgcn_wmma_f32_16x16x4_f32(false, a, false, b,
                                                (short)0, d, false, false);
    }
#pragma unroll
    for (int v = 0; v < 8; ++v) {
      const int feat = 16 * m + v + 8 * half;
      float val = d[v] + (feat < 20 ? b2[feat] : 0.0f);
      l2[feat * 16 + col] = val > 0.0f ? val : 0.0f;
    }
  }
  __syncthreads();

  // ---- Layer 3: 1x20, 1 M-tile x 5 K-tiles ----
  v8f d3 = {};
#pragma unroll
  for (int k = 0; k < 5; ++k) {
    const int kb = 4 * k + 2 * half;
    v2f a, b;
    a.x = (col == 0 && kb     < 20) ? W3[kb]     : 0.0f;
    a.y = (col == 0 && kb + 1 < 20) ? W3[kb + 1] : 0.0f;
    b.x = l2[(kb)     * 16 + col];  // kb max = 19 < 32 padded rows
    b.y = l2[(kb + 1) * 16 + col];
    d3 = __builtin_amdgcn_wmma_f32_16x16x4_f32(false, a, false, b,
                                               (short)0, d3, false, false);
  }

  // D row M=0 lives in d3[0] on lanes 0..15.
  const int nn = gbase + col;
  if (half == 0 && nn < nNodes) out[nn] = d3[0] + b3[0];
}

// ---------------- Host-side launcher ----------------
extern "C" void kernel_launch(void* const* d_in, const int* in_sizes, int n_in,
                              void* d_out, int out_size, void* d_ws, size_t ws_size,
                              hipStream_t stream) {
  const float* vertex     = (const float*)d_in[0];
  const int*   edge_index = (const int*)d_in[1];   // [2, E]
  const float* edge_attr  = (const float*)d_in[2]; // [E, 3]
  const float* W1 = (const float*)d_in[3];
  const float* b1 = (const float*)d_in[4];
  const float* W2 = (const float*)d_in[5];
  const float* b2 = (const float*)d_in[6];
  const float* W3 = (const float*)d_in[7];
  const float* b3 = (const float*)d_in[8];
  float* out = (float*)d_out;
  float* agg = (float*)d_ws;  // [N, 3] scratch accumulator

  const int nNodes = in_sizes[0] / 2;
  const int nEdges = in_sizes[2] / 3;
  const int* edge_dst = edge_index + nEdges;  // row 1 = targets

  const int aggElems = nNodes * 3;
  zero_agg_kernel<<<(aggElems + 255) / 256, 256, 0, stream>>>(agg, aggElems);
  edge_scatter_kernel<<<(nEdges + 255) / 256, 256, 0, stream>>>(
      edge_attr, edge_dst, agg, nEdges);

  const int nBlocks = (nNodes + 127) / 128;  // 8 waves/block, 16 nodes/wave
  mlp_wmma_kernel<<<nBlocks, 256, 0, stream>>>(vertex, agg, W1, b1, W2, b2,
                                               W3, b3, out, nNodes);
}